// SmoothnessLoss_52673478918525
// MI455X (gfx1250) — compile-verified
//
#include <hip/hip_runtime.h>
#include <math.h>

// Problem constants (fixed by the reference).
#define N_PTS   8192
#define E_DIM   32
#define KNN     20
#define TILE_M  32      // rows per wave (two 16-row WMMA strips)
#define TILE_N  16      // cols per WMMA tile
#define TILES_PER_BATCH (N_PTS / TILE_M)   // 256

typedef __attribute__((ext_vector_type(2))) float v2f;
typedef __attribute__((ext_vector_type(8))) float v8f;

#if __has_builtin(__builtin_amdgcn_wmma_f32_16x16x4_f32)
#define HAVE_WMMA_F32X4 1
#else
#define HAVE_WMMA_F32X4 0
#endif

__global__ __launch_bounds__(32)
void SmoothnessLoss_knn_kernel(const float* __restrict__ points,
                               const float* __restrict__ embeddings,
                               float* __restrict__ out)
{
    const int b  = blockIdx.x >> 8;                 // 256 row-tiles per batch
    const int m0 = (blockIdx.x & (TILES_PER_BATCH - 1)) * TILE_M;
    const float* __restrict__ P = points     + (size_t)b * N_PTS * 3;
    const float* __restrict__ E = embeddings + (size_t)b * N_PTS * E_DIM;

    const int  lane = threadIdx.x;                  // 0..31, single wave32
    const int  r    = lane & 15;
    const bool hi   = lane >= 16;

    __shared__ float qp[TILE_M][3];                 // query points of this tile
    __shared__ float sqm[TILE_M];                   // |p_i|^2 per row
    __shared__ float d2T[TILE_M][TILE_N + 1];       // stride 17 -> bank-conflict free

    // Stage query points + row norms (one row per lane).
    {
        float x = P[(size_t)(m0 + lane) * 3 + 0];
        float y = P[(size_t)(m0 + lane) * 3 + 1];
        float z = P[(size_t)(m0 + lane) * 3 + 2];
        qp[lane][0] = x; qp[lane][1] = y; qp[lane][2] = z;
        sqm[lane]   = x * x + y * y + z * z;
    }
    __syncthreads();

#if HAVE_WMMA_F32X4
    // A operands per ISA layout (32-bit A 16x4): lanes 0-15 hold K=0,1 in
    // VGPR0/VGPR1; lanes 16-31 hold K=2,3 (K=3 zero-padded; points are 3-D).
    v2f a0, a1;
    if (!hi) {
        a0.x = qp[r][0];      a0.y = qp[r][1];
        a1.x = qp[16 + r][0]; a1.y = qp[16 + r][1];
    } else {
        a0.x = qp[r][2];      a0.y = 0.0f;
        a1.x = qp[16 + r][2]; a1.y = 0.0f;
    }
#endif

    // Per-lane top-KNN list over SQUARED distances (sqrt is monotonic, so
    // ordering — and therefore the neighbor indices — is identical to the
    // reference). Ascending, lower index wins ties (matches jax.lax.top_k).
    // ALL accesses are statically indexed so nd/ni live purely in VGPRs.
    float nd[KNN];
    int   ni[KNN];
#pragma unroll
    for (int q = 0; q < KNN; ++q) { nd[q] = 3.402823466e+38f; ni[q] = 0; }

    const int iRow = m0 + lane;

    for (int j0 = 0; j0 < N_PTS; j0 += TILE_N) {
        // Each lane loads the column point for col r (lo and hi halves load the
        // same 16 points; |p_j|^2 falls out for free).
        const float* pc = P + (size_t)(j0 + r) * 3;
        float px = pc[0], py = pc[1], pz = pc[2];
        float sqj = px * px + py * py + pz * pz;

        if (j0 + TILE_N < N_PTS)
            __builtin_prefetch(P + (size_t)(j0 + TILE_N + r) * 3, 0, 3);

#if HAVE_WMMA_F32X4
        // B operand (4x16): lanes 0-15 -> K=0,1 ; lanes 16-31 -> K=2,3(pad).
        v2f bb; bb.x = hi ? pz : px; bb.y = hi ? 0.0f : py;
        v8f c0 = {}; v8f c1 = {};
        // (neg_a, A, neg_b, B, c_mod, C, reuse_a, reuse_b)
        c0 = __builtin_amdgcn_wmma_f32_16x16x4_f32(false, a0, false, bb,
                                                   (short)0, c0, false, false);
        c1 = __builtin_amdgcn_wmma_f32_16x16x4_f32(false, a1, false, bb,
                                                   (short)0, c1, false, false);
#endif
#pragma unroll
        for (int cc = 0; cc < 8; ++cc) {
            const int rr = cc + (hi ? 8 : 0);   // C layout: M = cc + 8*(lane>=16)
#if HAVE_WMMA_F32X4
            float dot0 = c0[cc];
            float dot1 = c1[cc];
#else
            float dot0 = qp[rr][0] * px + qp[rr][1] * py + qp[rr][2] * pz;
            float dot1 = qp[16 + rr][0] * px + qp[16 + rr][1] * py + qp[16 + rr][2] * pz;
#endif
            float d2a = sqm[rr]      + sqj - 2.0f * dot0;
            float d2b = sqm[16 + rr] + sqj - 2.0f * dot1;
            d2T[rr][r]      = fmaxf(d2a, 0.0f);   // no sqrt: ordering-equivalent
            d2T[16 + rr][r] = fmaxf(d2b, 0.0f);
        }
        __syncthreads();

        // Selection: lane owns row `lane`; scan 16 candidates. NOT unrolled
        // (one copy of the insert block keeps code inside the I-cache); the
        // insert itself is a branch-free predicated shift with static indices.
#pragma unroll 1
        for (int c = 0; c < TILE_N; ++c) {
            const int   j = j0 + c;
            const float d = d2T[lane][c];
            if (d < nd[KNN - 1] && j != iRow) {
#pragma unroll
                for (int p = KNN - 1; p > 0; --p) {
                    const bool gp  = nd[p]     > d;
                    const bool gpm = nd[p - 1] > d;
                    nd[p] = gp ? (gpm ? nd[p - 1] : d) : nd[p];
                    ni[p] = gp ? (gpm ? ni[p - 1] : j) : ni[p];
                }
                if (nd[0] > d) { nd[0] = d; ni[0] = j; }
            }
        }
        __syncthreads();
    }

    // Embedding smoothness: mean_q || E[nbr_q] - E[i] ||_2  (32 channels).
    // Fully unrolled so ni[q] stays statically indexed (no scratch).
    const float4* ei4 = (const float4*)(E + (size_t)iRow * E_DIM);
    float4 e[8];
#pragma unroll
    for (int c = 0; c < 8; ++c) e[c] = ei4[c];

    float acc = 0.0f;
#pragma unroll
    for (int q = 0; q < KNN; ++q) {
        const float4* en4 = (const float4*)(E + (size_t)ni[q] * E_DIM);
        float s = 0.0f;
#pragma unroll
        for (int c = 0; c < 8; ++c) {
            float4 v = en4[c];
            float dx = v.x - e[c].x, dy = v.y - e[c].y;
            float dz = v.z - e[c].z, dw = v.w - e[c].w;
            s = fmaf(dx, dx, fmaf(dy, dy, fmaf(dz, dz, fmaf(dw, dw, s))));
        }
        acc += sqrtf(s);   // 20 precise sqrts per thread, outside hot loop
    }

    // Global mean over (2 batches x 8192 rows x 20 neighbors).
    atomicAdd(out, acc * (1.0f / (2.0f * (float)N_PTS * (float)KNN)));
}

extern "C" void kernel_launch(void* const* d_in, const int* in_sizes, int n_in,
                              void* d_out, int out_size, void* d_ws, size_t ws_size,
                              hipStream_t stream) {
    const float* points     = (const float*)d_in[0];   // (2, 8192, 3)  fp32
    const float* embeddings = (const float*)d_in[1];   // (2, 8192, 32) fp32
    float* out = (float*)d_out;                        // scalar fp32

    hipMemsetAsync(out, 0, sizeof(float), stream);     // capturable memset node

    dim3 grid(2 * TILES_PER_BATCH);                    // 512 waves total
    dim3 block(32);                                    // one wave32 per block
    hipLaunchKernelGGL(SmoothnessLoss_knn_kernel, grid, block, 0, stream,
                       points, embeddings, out);
}